// Subsampling_Layer_26362509263654
// MI455X (gfx1250) — compile-verified
//
#include <hip/hip_runtime.h>
#include <math.h>

typedef __attribute__((ext_vector_type(2))) float v2f;
typedef __attribute__((ext_vector_type(8))) float v8f;

#define NN 320
#define M_REAL 6008
#define MP 6016            // M padded to multiple of 16
#define NCH 8
#define PLANE ((size_t)MP * NN)   // 1,925,120 floats per table plane

// ws layout (floats):
//   Axr  @ 0        Axi  @ 1*PLANE     (row-major [m][x] -> adjoint A operand)
//   Ayr  @ 2*PLANE  Ayi  @ 3*PLANE     (row-major [m][x])
//   AxrT @ 4*PLANE  AxiT @ 5*PLANE     (transposed [x][m] -> forward A operand)
//   ksp  @ 6*PLANE  (NCH*MP*2 floats)   total ~46.6 MB

// ---------------------------------------------------------------- kernel 1
__global__ __launch_bounds__(256)
void build_tables(const float* __restrict__ traj, float* __restrict__ ws) {
    size_t idx = (size_t)blockIdx.x * blockDim.x + threadIdx.x;
    if (idx >= PLANE) return;
    int m = (int)(idx / NN);
    int x = (int)(idx - (size_t)m * NN);
    float axr = 0.f, axi = 0.f, ayr = 0.f, ayi = 0.f;
    if (m < M_REAL) {
        float kx = traj[2 * m];
        float ky = traj[2 * m + 1];
        float g  = (float)(x - NN / 2);
        float w  = -2.0f * 3.14159265358979323846f / (float)NN;
        sincosf(w * kx * g, &axi, &axr);
        sincosf(w * ky * g, &ayi, &ayr);
    }
    ws[idx]             = axr;                   // Axr[m][x]
    ws[PLANE + idx]     = axi;                   // Axi[m][x]
    ws[2 * PLANE + idx] = ayr;                   // Ayr[m][x]
    ws[3 * PLANE + idx] = ayi;                   // Ayi[m][x]
    ws[4 * PLANE + (size_t)x * MP + m] = axr;    // AxrT[x][m]
    ws[5 * PLANE + (size_t)x * MP + m] = axi;    // AxiT[x][m]
}

// ---------------------------------------------------------------- kernel 2
// ksp[c,m] = sum_x sum_y Ax[m,x] * img[c,x,y] * Ay[m,y]
// block = (m-tile of 16, channel c), 4 waves; wave owns y-tiles {w, w+4, ...}
__global__ __launch_bounds__(128)
void forward_ksp(const float* __restrict__ img, const float* __restrict__ ws,
                 float* __restrict__ ksp) {
    const float* Ayr  = ws + 2 * PLANE;
    const float* Ayi  = ws + 3 * PLANE;
    const float* AxrT = ws + 4 * PLANE;
    const float* AxiT = ws + 5 * PLANE;

    const int m0  = blockIdx.x * 16;
    const int c   = blockIdx.y;
    const int tid = threadIdx.x;
    const int wave = tid >> 5;
    const int lane = tid & 31;
    const int j = lane & 15;
    const int h = lane >> 4;

    __shared__ float red[32];           // 16 m values * (re, im)
    if (tid < 32) red[tid] = 0.f;
    __syncthreads();

    float pr[8], pi[8];
#pragma unroll
    for (int v = 0; v < 8; ++v) { pr[v] = 0.f; pi[v] = 0.f; }

    // coalesced A loads: lane j indexes consecutive m in the transposed planes
    const float* aTr = AxrT + (m0 + j);
    const float* aTi = AxiT + (m0 + j);

    for (int yt = wave; yt < NN / 16; yt += 4) {
        const int y0 = yt * 16;
        v8f Prr = {}, Pii = {}, Pri = {}, Pir = {};
        for (int kk = 0; kk < NN / 4; ++kk) {
            const int xb = kk * 4 + 2 * h;              // K rows for this half-wave
            v2f Ar, Ai;
            Ar[0] = aTr[(size_t)(xb    ) * MP];         // Ax[m0+j][xb]
            Ar[1] = aTr[(size_t)(xb + 1) * MP];         // Ax[m0+j][xb+1]
            Ai[0] = aTi[(size_t)(xb    ) * MP];
            Ai[1] = aTi[(size_t)(xb + 1) * MP];
            // B[2h+r][j] = img[c][xb+r][y0+j]  (interleaved complex, coalesced)
            v2f c0 = *(const v2f*)(img + (((size_t)c * NN + xb    ) * NN + (y0 + j)) * 2);
            v2f c1 = *(const v2f*)(img + (((size_t)c * NN + xb + 1) * NN + (y0 + j)) * 2);
            v2f Br; Br[0] = c0[0]; Br[1] = c1[0];
            v2f Bi; Bi[0] = c0[1]; Bi[1] = c1[1];
            Prr = __builtin_amdgcn_wmma_f32_16x16x4_f32(false, Ar, false, Br, (short)0, Prr, false, false);
            Pii = __builtin_amdgcn_wmma_f32_16x16x4_f32(false, Ai, false, Bi, (short)0, Pii, false, false);
            Pri = __builtin_amdgcn_wmma_f32_16x16x4_f32(false, Ar, false, Bi, (short)0, Pri, false, false);
            Pir = __builtin_amdgcn_wmma_f32_16x16x4_f32(false, Ai, false, Br, (short)0, Pir, false, false);
        }
        // t[m][y] = (Prr - Pii) + i(Pri + Pir); multiply by Ay[m][y], reduce over y
#pragma unroll
        for (int v = 0; v < 8; ++v) {
            const int m = m0 + v + 8 * h;
            const int y = y0 + j;
            const float tr = Prr[v] - Pii[v];
            const float ti = Pri[v] + Pir[v];
            const float ayr = Ayr[(size_t)m * NN + y];
            const float ayi = Ayi[(size_t)m * NN + y];
            pr[v] += tr * ayr - ti * ayi;
            pi[v] += tr * ayi + ti * ayr;
        }
    }
    // reduce across the 16 j-lanes (xor masks < 16 stay within each half)
#pragma unroll
    for (int v = 0; v < 8; ++v) {
#pragma unroll
        for (int mask = 1; mask < 16; mask <<= 1) {
            pr[v] += __shfl_xor(pr[v], mask, 32);
            pi[v] += __shfl_xor(pi[v], mask, 32);
        }
    }
    if (j == 0) {
#pragma unroll
        for (int v = 0; v < 8; ++v) {
            atomicAdd(&red[(v + 8 * h) * 2 + 0], pr[v]);
            atomicAdd(&red[(v + 8 * h) * 2 + 1], pi[v]);
        }
    }
    __syncthreads();
    if (tid < 16) {
        size_t o = ((size_t)c * MP + m0 + tid) * 2;
        ksp[o]     = red[tid * 2];
        ksp[o + 1] = red[tid * 2 + 1];
    }
}

// ---------------------------------------------------------------- kernel 3
// out[c,x,w] = sum_m conj(Ax[m,x]) * ksp[c,m] * conj(Ay[m,w])
// block = (x-tile, w-tile, c), 4 waves split the K=MP loop, LDS reduce.
__global__ __launch_bounds__(128)
void adjoint_out(const float* __restrict__ ws, const float* __restrict__ ksp,
                 float* __restrict__ out) {
    const float* Axr = ws;                // row-major [m][x] -> coalesced A loads
    const float* Axi = ws + PLANE;
    const float* Ayr = ws + 2 * PLANE;
    const float* Ayi = ws + 3 * PLANE;

    const int x0  = blockIdx.x * 16;
    const int w0  = blockIdx.y * 16;
    const int c   = blockIdx.z;
    const int tid = threadIdx.x;
    const int wave = tid >> 5;
    const int lane = tid & 31;
    const int j = lane & 15;
    const int h = lane >> 4;

    __shared__ float acc[16 * 16 * 2];
    for (int i = tid; i < 512; i += 128) acc[i] = 0.f;
    __syncthreads();

    v8f Prr = {}, Pii = {}, Pri = {}, Pir = {};
    const float* aXr  = Axr + (x0 + j);           // lane j -> consecutive x
    const float* aXi  = Axi + (x0 + j);
    const float* aYr  = Ayr + (w0 + j);
    const float* aYi  = Ayi + (w0 + j);
    const float* kspC = ksp + (size_t)c * MP * 2;

    const int steps = MP / 16;                    // 376 k-steps per wave
    for (int kk = wave * steps; kk < (wave + 1) * steps; ++kk) {
        const int mb = kk * 4 + 2 * h;
        v2f Ar, Ai;
        Ar[0] = aXr[(size_t)(mb    ) * NN];       // Ax[mb  ][x0+j]
        Ar[1] = aXr[(size_t)(mb + 1) * NN];       // Ax[mb+1][x0+j]
        Ai[0] = aXi[(size_t)(mb    ) * NN];
        Ai[1] = aXi[(size_t)(mb + 1) * NN];
        v2f Gr, Gi;
#pragma unroll
        for (int r = 0; r < 2; ++r) {
            const int m = mb + r;
            v2f k2 = *(const v2f*)(kspC + (size_t)m * 2);   // (kr, ki)
            const float ayr = aYr[(size_t)m * NN];
            const float ayi = aYi[(size_t)m * NN];
            // G = ksp * conj(Ay)
            Gr[r] = k2[0] * ayr + k2[1] * ayi;
            Gi[r] = k2[1] * ayr - k2[0] * ayi;
        }
        Prr = __builtin_amdgcn_wmma_f32_16x16x4_f32(false, Ar, false, Gr, (short)0, Prr, false, false);
        Pii = __builtin_amdgcn_wmma_f32_16x16x4_f32(false, Ai, false, Gi, (short)0, Pii, false, false);
        Pri = __builtin_amdgcn_wmma_f32_16x16x4_f32(false, Ar, false, Gi, (short)0, Pri, false, false);
        Pir = __builtin_amdgcn_wmma_f32_16x16x4_f32(false, Ai, false, Gr, (short)0, Pir, false, false);
    }
    // conj(A): C = (ar - i*ai)(gr + i*gi) -> Cr = Prr + Pii, Ci = Pri - Pir
#pragma unroll
    for (int v = 0; v < 8; ++v) {
        const int row = v + 8 * h;
        atomicAdd(&acc[(row * 16 + j) * 2 + 0], Prr[v] + Pii[v]);
        atomicAdd(&acc[(row * 16 + j) * 2 + 1], Pri[v] - Pir[v]);
    }
    __syncthreads();
    for (int i = tid; i < 256; i += 128) {
        const int xr = i >> 4;
        const int wc = i & 15;
        size_t o = (((size_t)c * NN + x0 + xr) * NN + (w0 + wc)) * 2;
        out[o]     = acc[i * 2];
        out[o + 1] = acc[i * 2 + 1];
    }
}

// ---------------------------------------------------------------- launch
extern "C" void kernel_launch(void* const* d_in, const int* in_sizes, int n_in,
                              void* d_out, int out_size, void* d_ws, size_t ws_size,
                              hipStream_t stream) {
    (void)in_sizes; (void)n_in; (void)out_size; (void)ws_size;
    const float* img  = (const float*)d_in[0];   // (1,8,320,320,2) f32
    const float* traj = (const float*)d_in[1];   // (4,1502,2) f32
    float* ws  = (float*)d_ws;
    float* out = (float*)d_out;
    float* ksp = ws + 6 * PLANE;

    build_tables<<<dim3((unsigned)((PLANE + 255) / 256)), dim3(256), 0, stream>>>(traj, ws);
    forward_ksp<<<dim3(MP / 16, NCH), dim3(128), 0, stream>>>(img, ws, ksp);
    adjoint_out<<<dim3(NN / 16, NN / 16, NCH), dim3(128), 0, stream>>>(ws, ksp, out);
}